// GINWithJK_87909390614644
// MI455X (gfx1250) — compile-verified
//
#include <hip/hip_runtime.h>
#include <math.h>

typedef __attribute__((ext_vector_type(2))) float v2f;
typedef __attribute__((ext_vector_type(8))) float v8f;

#define H 64
#define SBUF 68           // padded LDS row stride (floats): 4m banks -> conflict-free A frags
#define BN_EPS 1e-5f

// ---------------------------------------------------------------------------
// One GEMM stage: D(16x64 strip) = A(rows row0..row0+15 of LDS tile) @ W(64x64 LDS)
// f32 WMMA 16x16x4, 16 K-steps. Wave handles 4 N-tiles of 16 cols.
// A layout: lane<16 -> K = kb,kb+1 ; lane>=16 -> K = kb+2,kb+3 (per CDNA5 ISA 7.12.2)
// ---------------------------------------------------------------------------
__device__ __forceinline__ void gemm_stage(const float* buf, const float* w,
                                           int row0, int lane, v8f* acc) {
  const int m  = lane & 15;
  const int kh = (lane >> 4) << 1;
  const float* arow = buf + (row0 + m) * SBUF;
#pragma unroll
  for (int k4 = 0; k4 < 16; ++k4) {
    const int kb = k4 * 4 + kh;
    v2f a;
    a.x = arow[kb];
    a.y = arow[kb + 1];
#pragma unroll
    for (int nt = 0; nt < 4; ++nt) {
      v2f b;
      b.x = w[kb * H + nt * 16 + m];
      b.y = w[(kb + 1) * H + nt * 16 + m];
      acc[nt] = __builtin_amdgcn_wmma_f32_16x16x4_f32(
          false, a, false, b, (short)0, acc[nt], false, false);
    }
  }
}

// epilogue: bias -> ReLU -> BatchNorm(eval) -> global store (row-guarded)
__device__ __forceinline__ void epilogue_store(
    const v8f* acc, const float* __restrict__ bia, const float* __restrict__ gg,
    const float* __restrict__ bb, const float* __restrict__ mm,
    const float* __restrict__ vv, float* __restrict__ out, long out_stride,
    long grow0, int m, long N) {
#pragma unroll
  for (int nt = 0; nt < 4; ++nt) {
    const int c = nt * 16 + m;
    const float bi = bia[c];
    const float sc = gg[c] * rsqrtf(vv[c] + BN_EPS);
    const float sh = bb[c] - mm[c] * sc;
#pragma unroll
    for (int j = 0; j < 8; ++j) {
      const long gr = grow0 + j;
      if (gr < N) {
        const float v = fmaxf(acc[nt][j] + bi, 0.0f);
        out[gr * out_stride + c] = v * sc + sh;
      }
    }
  }
}

// ---------------------------------------------------------------------------
// Fused GIN MLP: (optionally z=(1+eps)h+agg) -> GEMM1+ReLU+BN -> GEMM2+ReLU+BN
// TWO=false: single GEMM using the "1" parameter slots (used for layer 0 tail).
// Block = 128 threads (4 wave32), tile = 64 rows x 64 cols.
// ---------------------------------------------------------------------------
template <bool TWO>
__global__ __launch_bounds__(128) void gin_mlp_kernel(
    const float* __restrict__ hin, long hin_stride,
    const float* __restrict__ agg, const float* __restrict__ eps_ptr,
    const float* __restrict__ W1g, const float* __restrict__ b1,
    const float* __restrict__ g1, const float* __restrict__ be1,
    const float* __restrict__ m1, const float* __restrict__ v1,
    const float* __restrict__ W2g, const float* __restrict__ b2,
    const float* __restrict__ g2, const float* __restrict__ be2,
    const float* __restrict__ m2, const float* __restrict__ v2,
    float* __restrict__ out, long out_stride, long N) {
  __shared__ float zbuf[64 * SBUF];
  __shared__ float w1s[H * H];
  __shared__ float w2s[TWO ? H * H : 4];

  const int tid = threadIdx.x;
  const long rb = (long)blockIdx.x * 64;
  const float opeps = TWO ? (1.0f + eps_ptr[0]) : 1.0f;

  // stage activations (z) and weights into LDS
  for (int idx = tid; idx < 64 * H; idx += 128) {
    const int r = idx >> 6, c = idx & 63;
    const long gr = rb + r;
    float v = 0.0f;
    if (gr < N) {
      v = hin[gr * hin_stride + c];
      if (TWO) v = opeps * v + agg[gr * H + c];
    }
    zbuf[r * SBUF + c] = v;
  }
  for (int idx = tid; idx < H * H; idx += 128) w1s[idx] = W1g[idx];
  if (TWO)
    for (int idx = tid; idx < H * H; idx += 128) w2s[idx] = W2g[idx];
  __syncthreads();

  const int lane = tid & 31;
  const int wid  = tid >> 5;
  const int row0 = wid * 16;
  const int m    = lane & 15;
  const int rup  = (lane >> 4) * 8;

  v8f acc[4] = {};
  gemm_stage(zbuf, w1s, row0, lane, acc);

  if (TWO) {
    __syncthreads();  // everyone done reading zbuf
    // ReLU + BN1, write t back into zbuf
#pragma unroll
    for (int nt = 0; nt < 4; ++nt) {
      const int c = nt * 16 + m;
      const float bi = b1[c];
      const float sc = g1[c] * rsqrtf(v1[c] + BN_EPS);
      const float sh = be1[c] - m1[c] * sc;
#pragma unroll
      for (int j = 0; j < 8; ++j) {
        const float v = fmaxf(acc[nt][j] + bi, 0.0f);
        zbuf[(row0 + rup + j) * SBUF + c] = v * sc + sh;
      }
    }
    __syncthreads();
    v8f acc2[4] = {};
    gemm_stage(zbuf, w2s, row0, lane, acc2);
    epilogue_store(acc2, b2, g2, be2, m2, v2, out, out_stride,
                   rb + row0 + rup, m, N);
  } else {
    epilogue_store(acc, b1, g1, be1, m1, v1, out, out_stride,
                   rb + row0 + rup, m, N);
  }
}

// ---------------------------------------------------------------------------
// Layer-0 helpers (input dim = 1)
// ---------------------------------------------------------------------------
__global__ void scatter0_kernel(const float* __restrict__ x,
                                const int* __restrict__ src,
                                const int* __restrict__ dst,
                                float* __restrict__ agg0, long E) {
  const long e = (long)blockIdx.x * blockDim.x + threadIdx.x;
  if (e < E) atomicAdd(&agg0[dst[e]], x[src[e]]);
}

__global__ void layer0_kernel(const float* __restrict__ x,
                              const float* __restrict__ agg0,
                              const float* __restrict__ eps_ptr,
                              const float* __restrict__ W1f,
                              const float* __restrict__ b1,
                              const float* __restrict__ g1,
                              const float* __restrict__ be1,
                              const float* __restrict__ m1,
                              const float* __restrict__ v1,
                              float* __restrict__ t0, long N) {
  const long tid = (long)blockIdx.x * blockDim.x + threadIdx.x;
  if (tid >= N * H) return;
  const long n = tid >> 6;
  const int c = (int)(tid & 63);
  const float z = (1.0f + eps_ptr[0]) * x[n] + agg0[n];
  const float h = fmaxf(z * W1f[c] + b1[c], 0.0f);
  const float sc = g1[c] * rsqrtf(v1[c] + BN_EPS);
  t0[tid] = (h - m1[c]) * sc + be1[c];
}

// ---------------------------------------------------------------------------
// Edge scatter for H=64 layers: 32 lanes/edge, 2 channels/lane
// ---------------------------------------------------------------------------
__global__ void scatter_kernel(const float* __restrict__ h, long h_stride,
                               const int* __restrict__ src,
                               const int* __restrict__ dst,
                               float* __restrict__ agg, long E) {
  const long tid = (long)blockIdx.x * blockDim.x + threadIdx.x;
  if (tid >= E * 32) return;
  const long e = tid >> 5;
  const int c = (int)(tid & 31) * 2;
  const long s = src[e], d = dst[e];
  const float* hp = h + s * h_stride + c;
  const float a0 = hp[0], a1 = hp[1];
  atomicAdd(&agg[d * H + c], a0);
  atomicAdd(&agg[d * H + c + 1], a1);
}

// ---------------------------------------------------------------------------
// Mean-pool over batch (sums + counts via atomics)
// ---------------------------------------------------------------------------
__global__ void pool_kernel(const float* __restrict__ jk,
                            const int* __restrict__ batch,
                            float* __restrict__ pooled, long N) {
  const long tid = (long)blockIdx.x * blockDim.x + threadIdx.x;
  if (tid >= N * 128) return;
  const long n = tid >> 7;
  const int c = (int)(tid & 127) * 4;
  const long g = batch[n];
  const float* p = jk + n * 512 + c;
  float* q = pooled + g * 512 + c;
  atomicAdd(q + 0, p[0]);
  atomicAdd(q + 1, p[1]);
  atomicAdd(q + 2, p[2]);
  atomicAdd(q + 3, p[3]);
}

__global__ void count_kernel(const int* __restrict__ batch,
                             float* __restrict__ counts, long N) {
  const long n = (long)blockIdx.x * blockDim.x + threadIdx.x;
  if (n < N) atomicAdd(&counts[batch[n]], 1.0f);
}

// ---------------------------------------------------------------------------
// Head: mean, lin1+ReLU, lin2, log_softmax. One block (64 thr) per graph.
// ---------------------------------------------------------------------------
__global__ __launch_bounds__(64) void head_kernel(
    const float* __restrict__ pooled, const float* __restrict__ counts,
    const float* __restrict__ lin1_w, const float* __restrict__ lin1_b,
    const float* __restrict__ lin2_w, const float* __restrict__ lin2_b,
    float* __restrict__ out) {
  __shared__ float row[512];
  __shared__ float hid[64];
  __shared__ float logits[3];
  const int g = blockIdx.x;
  const int t = threadIdx.x;
  const float inv = 1.0f / fmaxf(counts[g], 1.0f);
  for (int i = t; i < 512; i += 64) row[i] = pooled[(long)g * 512 + i] * inv;
  __syncthreads();
  float a = lin1_b[t];
  for (int k = 0; k < 512; ++k) a = fmaf(row[k], lin1_w[k * 64 + t], a);
  hid[t] = fmaxf(a, 0.0f);
  __syncthreads();
  if (t < 3) {
    float l = lin2_b[t];
    for (int k = 0; k < 64; ++k) l = fmaf(hid[k], lin2_w[k * 3 + t], l);
    logits[t] = l;
  }
  __syncthreads();
  if (t == 0) {
    const float mx = fmaxf(logits[0], fmaxf(logits[1], logits[2]));
    const float s = expf(logits[0] - mx) + expf(logits[1] - mx) + expf(logits[2] - mx);
    const float ls = mx + logf(s);
    out[(long)g * 3 + 0] = logits[0] - ls;
    out[(long)g * 3 + 1] = logits[1] - ls;
    out[(long)g * 3 + 2] = logits[2] - ls;
  }
}

// ---------------------------------------------------------------------------
extern "C" void kernel_launch(void* const* d_in, const int* in_sizes, int n_in,
                              void* d_out, int out_size, void* d_ws, size_t ws_size,
                              hipStream_t stream) {
  const float* x        = (const float*)d_in[0];
  const int*   ei       = (const int*)d_in[1];
  const int*   batch    = (const int*)d_in[2];
  const float* eps      = (const float*)d_in[4];
  const float* W1_first = (const float*)d_in[5];
  const float* W1_rest  = (const float*)d_in[6];
  const float* b1       = (const float*)d_in[7];
  const float* g1       = (const float*)d_in[8];
  const float* be1      = (const float*)d_in[9];
  const float* m1       = (const float*)d_in[10];
  const float* v1       = (const float*)d_in[11];
  const float* W2       = (const float*)d_in[12];
  const float* b2       = (const float*)d_in[13];
  const float* g2       = (const float*)d_in[14];
  const float* be2      = (const float*)d_in[15];
  const float* m2       = (const float*)d_in[16];
  const float* v2       = (const float*)d_in[17];
  const float* lin1_w   = (const float*)d_in[18];
  const float* lin1_b   = (const float*)d_in[19];
  const float* lin2_w   = (const float*)d_in[20];
  const float* lin2_b   = (const float*)d_in[21];

  const long N = in_sizes[0];
  const long E = in_sizes[1] / 2;
  const long G = out_size / 3;
  const int* src = ei;
  const int* dst = ei + E;

  // workspace layout
  char* w = (char*)d_ws;
  float* jk     = (float*)w; w += (size_t)N * 512 * sizeof(float);
  float* agg    = (float*)w; w += (size_t)N * H * sizeof(float);   // doubles as t0
  float* agg0   = (float*)w; w += (size_t)N * sizeof(float);
  float* pooled = (float*)w; w += (size_t)G * 512 * sizeof(float);
  float* counts = (float*)w; w += (size_t)G * sizeof(float);
  float* t0 = agg;

  const int TB = 256;
  // ---- layer 0 ----
  hipMemsetAsync(agg0, 0, (size_t)N * sizeof(float), stream);
  scatter0_kernel<<<dim3((unsigned)((E + TB - 1) / TB)), TB, 0, stream>>>(
      x, src, dst, agg0, E);
  layer0_kernel<<<dim3((unsigned)((N * H + TB - 1) / TB)), TB, 0, stream>>>(
      x, agg0, eps, W1_first, b1, g1, be1, m1, v1, t0, N);
  gin_mlp_kernel<false><<<dim3((unsigned)((N + 63) / 64)), 128, 0, stream>>>(
      t0, H, nullptr, nullptr,
      W2, b2, g2, be2, m2, v2,                 // layer-0 second transform in slot 1
      nullptr, nullptr, nullptr, nullptr, nullptr, nullptr,
      jk, 512, N);

  // ---- layers 1..7 ----
  for (int i = 1; i < 8; ++i) {
    hipMemsetAsync(agg, 0, (size_t)N * H * sizeof(float), stream);
    scatter_kernel<<<dim3((unsigned)((E * 32 + TB - 1) / TB)), TB, 0, stream>>>(
        jk + (i - 1) * H, 512, src, dst, agg, E);
    gin_mlp_kernel<true><<<dim3((unsigned)((N + 63) / 64)), 128, 0, stream>>>(
        jk + (i - 1) * H, 512, agg, eps + i,
        W1_rest + (size_t)(i - 1) * H * H, b1 + i * H, g1 + i * H,
        be1 + i * H, m1 + i * H, v1 + i * H,
        W2 + (size_t)i * H * H, b2 + i * H, g2 + i * H,
        be2 + i * H, m2 + i * H, v2 + i * H,
        jk + i * H, 512, N);
  }

  // ---- pooling + head ----
  hipMemsetAsync(pooled, 0, (size_t)G * 512 * sizeof(float), stream);
  hipMemsetAsync(counts, 0, (size_t)G * sizeof(float), stream);
  pool_kernel<<<dim3((unsigned)((N * 128 + TB - 1) / TB)), TB, 0, stream>>>(
      jk, batch, pooled, N);
  count_kernel<<<dim3((unsigned)((N + TB - 1) / TB)), TB, 0, stream>>>(
      batch, counts, N);
  head_kernel<<<dim3((unsigned)G), 64, 0, stream>>>(
      pooled, counts, lin1_w, lin1_b, lin2_w, lin2_b, (float*)d_out);
}